// SplineActivation_70368744178508
// MI455X (gfx1250) — compile-verified
//
#include <hip/hip_runtime.h>
#include <stdint.h>

// SplineActivation: out[b,i] = sum_d B-spline-basis(x[b,i])_d * coeffs[i,d]
// GRID=5, ORDER=3, BASIS=7 -> only basis entries 0..3 are ever nonzero
// (right-clamped knot vector kills the x>=1 indicator and the pad entries),
// so only coeffs[i,0..3] matter.
//
// Memory-bound streaming kernel: ~67MB @ 23.3 TB/s ~= 2.9us floor.
// CDNA5 path used: global_load_async_to_lds_b128 (ASYNCcnt) to stage the
// per-tile coefficient slab into LDS, s_wait_asynccnt, then register-resident
// coeffs for a fully unrolled b128 streaming loop.

#define DIM    2048   // INPUT_DIM
#define BATCH  4096
#define TI     1024   // features per block (256 threads * 4)
#define ROWS   16     // batch rows per block
#define NT     256    // threads per block (8 waves on wave32)

// Exact replication of the Cox-de Boor recursion structure for the nonzero
// entries. Knots t = [-1,-.5,0,.5,1,1,1,1,1,1,1]. All /0.5 divisions are
// exact *2.0f; /1.5f replaced by *(2/3) (<=1 ulp).
__device__ __forceinline__ float spline4(float x, float c0, float c1, float c2, float c3) {
  const float xm1 = x + 1.0f;
  const float xp5 = x + 0.5f;
  const float xm5 = x - 0.5f;
  const float omx = 1.0f - x;
  const float hmx = 0.5f - x;

  const bool t0 = x >= -1.0f;
  const bool t1 = x >= -0.5f;
  const bool t2 = x >=  0.0f;
  const bool t3 = x >=  0.5f;
  const bool t4 = x >=  1.0f;
  const float b0 = (t0 && !t1) ? 1.0f : 0.0f;
  const float b1 = (t1 && !t2) ? 1.0f : 0.0f;
  const float b2 = (t2 && !t3) ? 1.0f : 0.0f;
  const float b3 = (t3 && !t4) ? 1.0f : 0.0f;

  // k = 1 (linear)
  const float n0 = 2.0f * xm1 * b0 + 2.0f * (-x) * b1;
  const float n1 = 2.0f * xp5 * b1 + 2.0f * hmx  * b2;
  const float n2 = 2.0f * x   * b2 + 2.0f * omx  * b3;
  const float n3 = 2.0f * xm5 * b3;

  // k = 2 (quadratic)
  const float m0 = xm1 * n0 + hmx * n1;
  const float m1 = xp5 * n1 + omx * n2;
  const float m2 = x   * n2 + 2.0f * omx * n3;
  const float m3 = 2.0f * xm5 * n3;

  // k = 3 (cubic)
  const float r15 = 2.0f / 3.0f;  // 1/1.5
  const float f0 = (xm1 * r15) * m0 + (omx * r15) * m1;
  const float f1 = (xp5 * r15) * m1 + omx * m2;
  const float f2 = x * m2 + 2.0f * omx * m3;
  const float f3 = 2.0f * xm5 * m3;

  return c0 * f0 + c1 * f1 + c2 * f2 + c3 * f3;
}

__global__ __launch_bounds__(NT) void SplineActivation_kernel(
    const float* __restrict__ x,
    const float* __restrict__ coeffs,
    float* __restrict__ out) {
  // Coefficient slab for this i-tile: TI features * 7 floats, contiguous
  // in memory (28 KB). Staged to LDS via CDNA5 async load-to-LDS.
  __shared__ __align__(16) float sc[TI * 7];

  const int iBase   = blockIdx.x * TI;
  const int rowBase = blockIdx.y * ROWS;

  const float* gsrc = coeffs + (size_t)iBase * 7;

  // TI*7 floats = 1792 float4 chunks; 256 threads * 7 chunks each.
  // Each lane issues async DMA: global -> LDS (tracked by ASYNCcnt).
#pragma unroll
  for (int k = 0; k < 7; ++k) {
    const int chunk = threadIdx.x + k * NT;
    // Low 32 bits of the generic (flat) shared pointer == LDS byte offset.
    unsigned lds_off = (unsigned)(uintptr_t)(sc + chunk * 4);
    const float* gp = gsrc + chunk * 4;
    asm volatile("global_load_async_to_lds_b128 %0, %1, off"
                 :: "v"(lds_off), "v"(gp)
                 : "memory");
  }
  // Each wave waits for its own async transfers, then cross-wave barrier.
  asm volatile("s_wait_asynccnt 0x0" ::: "memory");
  __syncthreads();

  // Hoist the 16 relevant coefficients (4 features x first 4 basis coeffs)
  // into registers once per block lifetime.
  const int li = threadIdx.x * 4;
  float c[4][4];
#pragma unroll
  for (int j = 0; j < 4; ++j) {
#pragma unroll
    for (int d = 0; d < 4; ++d) {
      c[j][d] = sc[(li + j) * 7 + d];
    }
  }

  const size_t colBase = (size_t)iBase + (size_t)li;

  // Streaming loop: b128 load of x, compute, b128 store of out.
#pragma unroll
  for (int r = 0; r < ROWS; ++r) {
    const size_t off = (size_t)(rowBase + r) * DIM + colBase;
    const float4 xv = *(const float4*)(x + off);
    float4 o;
    o.x = spline4(xv.x, c[0][0], c[0][1], c[0][2], c[0][3]);
    o.y = spline4(xv.y, c[1][0], c[1][1], c[1][2], c[1][3]);
    o.z = spline4(xv.z, c[2][0], c[2][1], c[2][2], c[2][3]);
    o.w = spline4(xv.w, c[3][0], c[3][1], c[3][2], c[3][3]);
    *(float4*)(out + off) = o;
  }
}

extern "C" void kernel_launch(void* const* d_in, const int* in_sizes, int n_in,
                              void* d_out, int out_size, void* d_ws, size_t ws_size,
                              hipStream_t stream) {
  const float* x      = (const float*)d_in[0];   // [BATCH, DIM] f32
  const float* coeffs = (const float*)d_in[1];   // [DIM, 7] f32
  float* out          = (float*)d_out;           // [BATCH, DIM] f32
  (void)in_sizes; (void)n_in; (void)out_size; (void)d_ws; (void)ws_size;

  dim3 grid(DIM / TI, BATCH / ROWS);  // (2, 256) blocks
  dim3 block(NT);                     // 256 threads = 8 wave32
  SplineActivation_kernel<<<grid, block, 0, stream>>>(x, coeffs, out);
}